// MyModel_61933428412962
// MI455X (gfx1250) — compile-verified
//
#include <hip/hip_runtime.h>

// ---------------------------------------------------------------------------
// 524288-point FFT = 16 x 16 x (16 x 16 x 8), radix-16 stages on f32 WMMA.
// Output: res[h,g,f] at p = 256h+16g+f, value X[f+16g+256h] as [[re,-im],[im,re]].
// Twiddles via hardware V_SIN/V_COS (argument in revolutions: exact j/M turns).
// ---------------------------------------------------------------------------

typedef float v2f __attribute__((ext_vector_type(2)));
typedef float v8f __attribute__((ext_vector_type(8)));

static __device__ __forceinline__ v8f wmma4(v2f a, v2f b, v8f c) {
  // D(16x16) += A(16x4) * B(4x16), f32.
  return __builtin_amdgcn_wmma_f32_16x16x4_f32(false, a, false, b, (short)0, c,
                                               false, false);
}

// exp(-2*pi*i*turn): one v_sin + one v_cos, no libm range reduction.
static __device__ __forceinline__ void cis_neg_turn(float turn, float& c,
                                                    float& s) {
  s = __builtin_amdgcn_sinf(-turn);
  c = __builtin_amdgcn_cosf(turn);  // cos is even
}

// Per-lane A-fragments of W16[m][k] = exp(-2*pi*i*m*k/16).
// f32 A 16x4 layout (ISA 7.12.2): lane l<16 -> row m=l, {K=4c+0, 4c+1};
// lane l>=16 -> row m=l-16, {K=4c+2, 4c+3}.
static __device__ __forceinline__ void make_w16(int lane, v2f wr[4], v2f wi[4],
                                                v2f wni[4]) {
  const int m = lane & 15;
  const int kb = (lane >> 4) * 2;
#pragma unroll
  for (int c = 0; c < 4; ++c) {
    const int k0 = 4 * c + kb;
    const float t0 = (float)((m * k0) & 15) * (1.0f / 16.0f);
    const float t1 = (float)((m * (k0 + 1)) & 15) * (1.0f / 16.0f);
    float c0, s0, c1, s1;
    cis_neg_turn(t0, c0, s0);
    cis_neg_turn(t1, c1, s1);
    wr[c].x = c0;  wr[c].y = c1;
    wi[c].x = s0;  wi[c].y = s1;
    wni[c].x = -s0; wni[c].y = -s1;
  }
}

// ---------------------------------------------------------------------------
// K1: radix-16 over x (real input), twiddle W_N^{f*(2048y+z)}, radix-16 over y,
// twiddle W_32768^{g*z}. One workgroup owns 16 consecutive z. Output ws[gf][z].
// ---------------------------------------------------------------------------
#define IN_STRIDE 264  // 16*16 + 8 pad: 2*264 mod 64 = 16 -> conflict-free K reads
#define A_STRIDE 260   // 8*260 mod 64 = 32 (D stores), +32 on K reads: disjoint banks

__global__ __launch_bounds__(256) void fft_k1_xy(const float* __restrict__ in,
                                                 float2* __restrict__ ws) {
  __shared__ float ldsIn[16 * IN_STRIDE];
  __shared__ float ldsAre[16 * A_STRIDE];
  __shared__ float ldsAim[16 * A_STRIDE];

  const int t = threadIdx.x;
  const int lane = t & 31;
  const int wave = t >> 5;
  const int Z0 = blockIdx.x * 16;

  // Cooperative load: thread (y = t>>4 & 15, z = t&15) walks x. Coalesced 64B/row.
  {
    const int y = (t >> 4) & 15;
    const int z = t & 15;
#pragma unroll
    for (int x = 0; x < 16; ++x)
      ldsIn[x * IN_STRIDE + y * 16 + z] = in[(x * 16 + y) * 2048 + Z0 + z];
  }
  __syncthreads();

  v2f wr[4], wi[4], wni[4];
  make_w16(lane, wr, wi, wni);

  const int n = lane & 15;
  const int kb = (lane >> 4) * 2;
  const int mh = (lane >> 4) * 8;

  // ---- Stage over x: A[f][y][z] (input is real: only 2 matmuls per tile) ----
#pragma unroll
  for (int yy = wave; yy < 16; yy += 8) {
    v8f cre = {}, cim = {};
    const float* bp = ldsIn + yy * 16;
#pragma unroll
    for (int c = 0; c < 4; ++c) {
      const int k0 = 4 * c + kb;
      v2f b;
      b.x = bp[k0 * IN_STRIDE + n];
      b.y = bp[(k0 + 1) * IN_STRIDE + n];
      cre = wmma4(wr[c], b, cre);
      cim = wmma4(wi[c], b, cim);
    }
    const int iz = yy * 2048 + Z0 + n;  // 2048*y + z
#pragma unroll
    for (int v = 0; v < 8; ++v) {
      const int f = v + mh;
      const float tw = (float)((f * iz) & 524287) * (1.0f / 524288.0f);
      float c, s;
      cis_neg_turn(tw, c, s);
      const float re = cre[v], im = cim[v];
      ldsAre[f * A_STRIDE + yy * 16 + n] = re * c - im * s;
      ldsAim[f * A_STRIDE + yy * 16 + n] = re * s + im * c;
    }
  }
  __syncthreads();

  // ---- Stage over y: B[g][f][z], full complex matmul (4 real WMMAs/chunk) ----
#pragma unroll
  for (int ff = wave; ff < 16; ff += 8) {
    v8f cre = {}, cim = {};
    const float* br = ldsAre + ff * A_STRIDE;
    const float* bi = ldsAim + ff * A_STRIDE;
#pragma unroll
    for (int c = 0; c < 4; ++c) {
      const int k0 = 4 * c + kb;
      v2f xr, xi;
      xr.x = br[k0 * 16 + n]; xr.y = br[(k0 + 1) * 16 + n];
      xi.x = bi[k0 * 16 + n]; xi.y = bi[(k0 + 1) * 16 + n];
      cre = wmma4(wr[c], xr, cre);
      cre = wmma4(wni[c], xi, cre);
      cim = wmma4(wi[c], xr, cim);
      cim = wmma4(wr[c], xi, cim);
    }
    const int z = Z0 + n;
#pragma unroll
    for (int v = 0; v < 8; ++v) {
      const int g = v + mh;
      const float tw = (float)((g * z) & 32767) * (1.0f / 32768.0f);
      float c, s;
      cis_neg_turn(tw, c, s);
      const float re = cre[v], im = cim[v];
      float2 o;
      o.x = re * c - im * s;
      o.y = re * s + im * c;
      ws[(g * 16 + ff) * 2048 + z] = o;  // 128B contiguous per half-wave
    }
  }
}

// ---------------------------------------------------------------------------
// K2: 2048-point FFT per gf row: radix-16 (z0) -> tw -> radix-16 (z1) -> tw ->
// radix-8 (z2). One workgroup per row; row lives in LDS. Output ws2[gf][h].
// ---------------------------------------------------------------------------
#define Z_STRIDE 140  // 128 + 12 pad: 2*140 mod 64 = 24, 8*140 mod 64 = 32

__global__ __launch_bounds__(256) void fft_k2_z(const float2* __restrict__ ws,
                                                float2* __restrict__ ws2) {
  __shared__ float zre[16 * Z_STRIDE], zim[16 * Z_STRIDE];
  __shared__ float bre[16 * Z_STRIDE], bim[16 * Z_STRIDE];

  const int t = threadIdx.x;
  const int lane = t & 31;
  const int wave = t >> 5;
  const int gf = blockIdx.x;

  // Load 2048 complex, layout [z0][c] with c = 8*z1 + z2 = z & 127.
#pragma unroll
  for (int i = 0; i < 8; ++i) {
    const int z = t + 256 * i;
    const float2 v = ws[gf * 2048 + z];
    zre[(z >> 7) * Z_STRIDE + (z & 127)] = v.x;
    zim[(z >> 7) * Z_STRIDE + (z & 127)] = v.y;
  }
  __syncthreads();

  v2f wr[4], wi[4], wni[4];
  make_w16(lane, wr, wi, wni);

  const int n = lane & 15;
  const int kb = (lane >> 4) * 2;
  const int mh = (lane >> 4) * 8;

  // ---- Stage z0: C1[h0][c] = sum_z0 zz[z0][c] W16^{z0 h0} * W2048^{h0 c} ----
  {
    const int col = wave * 16 + n;  // c = 8*z1 + z2
    v8f cre = {}, cim = {};
#pragma unroll
    for (int c = 0; c < 4; ++c) {
      const int k0 = 4 * c + kb;
      v2f xr, xi;
      xr.x = zre[k0 * Z_STRIDE + col]; xr.y = zre[(k0 + 1) * Z_STRIDE + col];
      xi.x = zim[k0 * Z_STRIDE + col]; xi.y = zim[(k0 + 1) * Z_STRIDE + col];
      cre = wmma4(wr[c], xr, cre);
      cre = wmma4(wni[c], xi, cre);
      cim = wmma4(wi[c], xr, cim);
      cim = wmma4(wr[c], xi, cim);
    }
#pragma unroll
    for (int v = 0; v < 8; ++v) {
      const int h0 = v + mh;
      const float tw = (float)((h0 * col) & 2047) * (1.0f / 2048.0f);
      float c, s;
      cis_neg_turn(tw, c, s);
      const float re = cre[v], im = cim[v];
      bre[h0 * Z_STRIDE + col] = re * c - im * s;
      bim[h0 * Z_STRIDE + col] = re * s + im * c;
    }
  }
  __syncthreads();

  // ---- Stage z1: C2[h1][h0*8+z2] = sum_z1 C1[h0][8z1+z2] W16^{z1 h1} * W128^{h1 z2}
  {
    const int col2 = wave * 16 + n;  // h0*8 + z2
    const int h0 = col2 >> 3, z2 = col2 & 7;
    const int base = h0 * Z_STRIDE + z2;
    v8f cre = {}, cim = {};
#pragma unroll
    for (int c = 0; c < 4; ++c) {
      const int k0 = 4 * c + kb;
      v2f xr, xi;
      xr.x = bre[base + k0 * 8]; xr.y = bre[base + (k0 + 1) * 8];
      xi.x = bim[base + k0 * 8]; xi.y = bim[base + (k0 + 1) * 8];
      cre = wmma4(wr[c], xr, cre);
      cre = wmma4(wni[c], xi, cre);
      cim = wmma4(wi[c], xr, cim);
      cim = wmma4(wr[c], xi, cim);
    }
#pragma unroll
    for (int v = 0; v < 8; ++v) {
      const int h1 = v + mh;
      const float tw = (float)((h1 * z2) & 127) * (1.0f / 128.0f);
      float c, s;
      cis_neg_turn(tw, c, s);
      const float re = cre[v], im = cim[v];
      zre[h1 * Z_STRIDE + col2] = re * c - im * s;  // reuse z-planes as C2
      zim[h1 * Z_STRIDE + col2] = re * s + im * c;
    }
  }
  __syncthreads();

  // ---- Stage z2: per-thread radix-8 over z2; batch (h0,h1) = t. ----
  {
    const int h0 = t & 15, h1 = t >> 4;
    float vr[8], vi[8];
#pragma unroll
    for (int z2 = 0; z2 < 8; ++z2) {
      vr[z2] = zre[h1 * Z_STRIDE + h0 * 8 + z2];
      vi[z2] = zim[h1 * Z_STRIDE + h0 * 8 + z2];
    }
    const float R = 0.70710678118654752440f;
    const float w8r[8] = {1.f, R, 0.f, -R, -1.f, -R, 0.f, R};
    const float w8i[8] = {0.f, -R, -1.f, -R, 0.f, R, 1.f, R};
#pragma unroll
    for (int h2 = 0; h2 < 8; ++h2) {
      float sr = 0.f, si = 0.f;
#pragma unroll
      for (int z2 = 0; z2 < 8; ++z2) {
        const int idx = (z2 * h2) & 7;
        sr += vr[z2] * w8r[idx] - vi[z2] * w8i[idx];
        si += vr[z2] * w8i[idx] + vi[z2] * w8r[idx];
      }
      ws2[gf * 2048 + (t + 256 * h2)] = make_float2(sr, si);  // coalesced
    }
  }
}

// ---------------------------------------------------------------------------
// K3: transpose ws2 (256 gf x 2048 h) -> out[p = 256h + gf] = [[re,-im],[im,re]].
// LDS-tiled so the 8 MB output store is fully coalesced (512B runs per h).
// ---------------------------------------------------------------------------
__global__ __launch_bounds__(256) void fft_k3_out(const float2* __restrict__ ws2,
                                                  float4* __restrict__ out) {
  __shared__ float2 tile[32][33];
  const int tx = threadIdx.x & 31;
  const int ty = threadIdx.x >> 5;
  const int h0 = blockIdx.x * 32;
  const int gf0 = blockIdx.y * 32;

#pragma unroll
  for (int i = 0; i < 4; ++i) {
    const int gf = ty + i * 8;
    tile[gf][tx] = ws2[(gf0 + gf) * 2048 + h0 + tx];
  }
  __syncthreads();
#pragma unroll
  for (int i = 0; i < 4; ++i) {
    const int hh = ty + i * 8;
    const float2 v = tile[tx][hh];
    out[(h0 + hh) * 256 + gf0 + tx] = make_float4(v.x, -v.y, v.y, v.x);
  }
}

// ---------------------------------------------------------------------------
extern "C" void kernel_launch(void* const* d_in, const int* in_sizes, int n_in,
                              void* d_out, int out_size, void* d_ws,
                              size_t ws_size, hipStream_t stream) {
  (void)in_sizes; (void)n_in; (void)out_size; (void)ws_size;
  const float* in = (const float*)d_in[0];
  float2* ws = (float2*)d_ws;            // 256*2048 complex = 4 MB
  float2* ws2 = ws + 256 * 2048;         // 256*2048 complex = 4 MB

  fft_k1_xy<<<128, 256, 0, stream>>>(in, ws);
  fft_k2_z<<<256, 256, 0, stream>>>(ws, ws2);
  fft_k3_out<<<dim3(2048 / 32, 256 / 32), 256, 0, stream>>>(ws2, (float4*)d_out);
}